// UnrolledProgressNet_31585189494791
// MI455X (gfx1250) — compile-verified
//
#include <hip/hip_runtime.h>
#include <math.h>

typedef __attribute__((ext_vector_type(2))) float v2f;
typedef __attribute__((ext_vector_type(8))) float v8f;

__device__ __forceinline__ float sigmoidf_(float x) { return 1.0f / (1.0f + expf(-x)); }

// ---------------------------------------------------------------------------
// Prep: pad+transpose W_spp (4096x90) -> Wb1 (96x4096), zero rows 90..95
// ---------------------------------------------------------------------------
__global__ void pad_transpose_wspp(const float* __restrict__ W, float* __restrict__ Wb) {
    int idx = blockIdx.x * 256 + threadIdx.x;   // 96*4096 threads
    int k = idx >> 12;          // / 4096
    int n = idx & 4095;
    Wb[idx] = (k < 90) ? W[n * 90 + k] : 0.0f;
}

// Transpose W (64x4096) -> Wb (4096x64)
__global__ void transpose_w64(const float* __restrict__ W, float* __restrict__ Wb) {
    int idx = blockIdx.x * 256 + threadIdx.x;   // 4096*64 threads
    int k = idx >> 6;
    int n = idx & 63;
    Wb[idx] = W[(size_t)n * 4096 + k];
}

// ---------------------------------------------------------------------------
// SPP: one block per (frame, channel). 192x192 -> 12x12 fine max grid -> levels
// Output layout per frame row (96 floats, 90 used):
//   [0,48)  level4: c*16 + i*4+j
//   [48,75) level3: 48 + c*9 + i*3+j
//   [75,87) level2: 75 + c*4 + i*2+j
//   [87,90) level1: 87 + c
// ---------------------------------------------------------------------------
__global__ __launch_bounds__(160) void spp_kernel(const float* __restrict__ frames,
                                                  float* __restrict__ sppP) {
    __shared__ float fine[144];
    int s = blockIdx.x / 3;
    int c = blockIdx.x % 3;
    int t = threadIdx.x;
    const float* base = frames + ((size_t)s * 3 + c) * (192 * 192);

    if (t < 144) {
        int br = t / 12, bc = t % 12;
        float m = -__builtin_inff();
        for (int r = 0; r < 16; r++) {
            const float4* p = (const float4*)(base + (size_t)(br * 16 + r) * 192 + bc * 16);
#pragma unroll
            for (int q = 0; q < 4; q++) {
                float4 v = p[q];
                m = fmaxf(m, fmaxf(fmaxf(v.x, v.y), fmaxf(v.z, v.w)));
            }
        }
        fine[t] = m;
    }
    __syncthreads();

    float* out = sppP + (size_t)s * 96;
    if (t < 16) {                    // level 4: 3x3 fine blocks each
        int i = t >> 2, j = t & 3;
        float m = -__builtin_inff();
        for (int di = 0; di < 3; di++)
            for (int dj = 0; dj < 3; dj++)
                m = fmaxf(m, fine[(3 * i + di) * 12 + (3 * j + dj)]);
        out[c * 16 + t] = m;
    } else if (t < 25) {             // level 3: 4x4 fine blocks each
        int u = t - 16, i = u / 3, j = u % 3;
        float m = -__builtin_inff();
        for (int di = 0; di < 4; di++)
            for (int dj = 0; dj < 4; dj++)
                m = fmaxf(m, fine[(4 * i + di) * 12 + (4 * j + dj)]);
        out[48 + c * 9 + u] = m;
    } else if (t < 29) {             // level 2: 6x6 fine blocks each
        int u = t - 25, i = u >> 1, j = u & 1;
        float m = -__builtin_inff();
        for (int di = 0; di < 6; di++)
            for (int dj = 0; dj < 6; dj++)
                m = fmaxf(m, fine[(6 * i + di) * 12 + (6 * j + dj)]);
        out[75 + c * 4 + u] = m;
    } else if (t == 29) {            // level 1: global max
        float m = -__builtin_inff();
        for (int q = 0; q < 144; q++) m = fmaxf(m, fine[q]);
        out[87 + c] = m;
    }
    if (c == 0 && t >= 30 && t < 36) out[90 + (t - 30)] = 0.0f;  // K-pad to 96
}

// ---------------------------------------------------------------------------
// fp32 WMMA GEMM: C = relu(A(M x K, lda) @ B(K x N, ldb) + bias), one wave per
// 16x16 tile, K stepped by 4 via V_WMMA_F32_16X16X4_F32.
// A frag: lane(lo)=M row, hi selects K pair {0,1}/{2,3}. B frag mirrors with N.
// C frag: VGPR r holds (M = r + 8*hi, N = lo).
// ---------------------------------------------------------------------------
__global__ __launch_bounds__(256) void gemm_relu_wmma(
    const float* __restrict__ A, const float* __restrict__ B,
    const float* __restrict__ bias, float* __restrict__ C,
    int numTiles, int tilesPerRowN, int K, int lda, int ldb, int ldc) {
    int wave = threadIdx.x >> 5;
    int tile = blockIdx.x * 8 + wave;
    if (tile >= numTiles) return;                 // wave-uniform: EXEC stays full
    int mt = tile / tilesPerRowN;
    int nt = tile % tilesPerRowN;
    int lane = threadIdx.x & 31;
    int lo = lane & 15, hi = lane >> 4;

    const float* arow = A + (size_t)(mt * 16 + lo) * lda;
    const float* bcol = B + (nt * 16 + lo);
    v8f acc = {0.f, 0.f, 0.f, 0.f, 0.f, 0.f, 0.f, 0.f};

    for (int k = 0; k < K; k += 4) {
        v2f a, b;
        a.x = arow[k + 2 * hi];
        a.y = arow[k + 2 * hi + 1];
        b.x = bcol[(size_t)(k + 2 * hi) * ldb];
        b.y = bcol[(size_t)(k + 2 * hi + 1) * ldb];
        acc = __builtin_amdgcn_wmma_f32_16x16x4_f32(false, a, false, b,
                                                    (short)0, acc, false, false);
    }

    int col = nt * 16 + lo;
    float bv = bias[col];
#pragma unroll
    for (int r = 0; r < 8; r++) {
        int row = mt * 16 + r + 8 * hi;
        float v = acc[r] + bv;
        C[(size_t)row * ldc + col] = v > 0.0f ? v : 0.0f;
    }
}

// ---------------------------------------------------------------------------
// Sequential LSTM scan (512 steps), single block, 256 threads.
// Cell2 weights + fused biases live in LDS (<64KB); cell1 weights stream from
// L2 (128KB resident in 192MB L2). Heads written directly to d_out.
// ---------------------------------------------------------------------------
__global__ __launch_bounds__(256) void lstm_kernel(
    const float* __restrict__ emb, const float* __restrict__ fct,
    const float* __restrict__ Wih1, const float* __restrict__ Whh1,
    const float* __restrict__ bih1, const float* __restrict__ bhh1,
    const float* __restrict__ Wih2, const float* __restrict__ Whh2,
    const float* __restrict__ bih2, const float* __restrict__ bhh2,
    const float* __restrict__ Wfc8, const float* __restrict__ bfc8,
    float* __restrict__ out) {
    __shared__ float sWih2[128 * 64];   // 32 KB
    __shared__ float sWhh2[128 * 32];   // 16 KB
    __shared__ float sb1[256], sb2[128];
    __shared__ float h1[64], c1[64], h2[32], c2[32], fh1[64], fh2[32];
    __shared__ float g1[256], g2[128];

    int t = threadIdx.x;
    for (int i = t; i < 128 * 64; i += 256) sWih2[i] = Wih2[i];
    for (int i = t; i < 128 * 32; i += 256) sWhh2[i] = Whh2[i];
    sb1[t] = bih1[t] + bhh1[t];
    if (t < 128) sb2[t] = bih2[t] + bhh2[t];
    if (t < 64) { h1[t] = 0.0f; c1[t] = 0.0f; }
    if (t < 32) { h2[t] = 0.0f; c2[t] = 0.0f; }
    __syncthreads();

    for (int s = 0; s < 512; s++) {
        const float* x  = emb + (size_t)s * 64;
        const float* fx = fct + (size_t)s * 64;

        // ---- main cell 1: g1 = Wih1@x + Whh1@h1 + b
        {
            float acc = sb1[t];
            const float* wi = Wih1 + (size_t)t * 64;
            const float* wh = Whh1 + (size_t)t * 64;
            for (int k = 0; k < 64; k++) acc += wi[k] * x[k] + wh[k] * h1[k];
            g1[t] = acc;
        }
        __syncthreads();
        if (t < 64) {
            float i_ = sigmoidf_(g1[t]);
            float f_ = sigmoidf_(g1[64 + t]);
            float gg = tanhf(g1[128 + t]);
            float o_ = sigmoidf_(g1[192 + t]);
            float cn = f_ * c1[t] + i_ * gg;
            c1[t] = cn;
            h1[t] = o_ * tanhf(cn);
        }
        __syncthreads();

        // ---- main cell 2
        if (t < 128) {
            float acc = sb2[t];
            const float* wi = sWih2 + t * 64;
            const float* wh = sWhh2 + t * 32;
            for (int k = 0; k < 64; k++) acc += wi[k] * h1[k];
            for (int k = 0; k < 32; k++) acc += wh[k] * h2[k];
            g2[t] = acc;
        }
        __syncthreads();
        if (t < 32) {
            float i_ = sigmoidf_(g2[t]);
            float f_ = sigmoidf_(g2[32 + t]);
            float gg = tanhf(g2[64 + t]);
            float o_ = sigmoidf_(g2[96 + t]);
            float cn = f_ * c2[t] + i_ * gg;
            c2[t] = cn;
            h2[t] = o_ * tanhf(cn);
        }
        __syncthreads();

        // ---- forecast cell 1 (uses updated h1, c1; state not kept)
        {
            float acc = sb1[t];
            const float* wi = Wih1 + (size_t)t * 64;
            const float* wh = Whh1 + (size_t)t * 64;
            for (int k = 0; k < 64; k++) acc += wi[k] * fx[k] + wh[k] * h1[k];
            g1[t] = acc;
        }
        __syncthreads();
        if (t < 64) {
            float i_ = sigmoidf_(g1[t]);
            float f_ = sigmoidf_(g1[64 + t]);
            float gg = tanhf(g1[128 + t]);
            float o_ = sigmoidf_(g1[192 + t]);
            float cn = f_ * c1[t] + i_ * gg;
            fh1[t] = o_ * tanhf(cn);
        }
        __syncthreads();

        // ---- forecast cell 2 (uses updated h2, c2)
        if (t < 128) {
            float acc = sb2[t];
            const float* wi = sWih2 + t * 64;
            const float* wh = sWhh2 + t * 32;
            for (int k = 0; k < 64; k++) acc += wi[k] * fh1[k];
            for (int k = 0; k < 32; k++) acc += wh[k] * h2[k];
            g2[t] = acc;
        }
        __syncthreads();
        if (t < 32) {
            float i_ = sigmoidf_(g2[t]);
            float f_ = sigmoidf_(g2[32 + t]);
            float gg = tanhf(g2[64 + t]);
            float o_ = sigmoidf_(g2[96 + t]);
            float cn = f_ * c2[t] + i_ * gg;
            fh2[t] = o_ * tanhf(cn);
        }
        __syncthreads();

        // ---- heads
        if (t == 0) {
            float a = bfc8[0];
            for (int j = 0; j < 32; j++) a += h2[j] * Wfc8[j];
            out[s] = sigmoidf_(a);
        }
        if (t == 32) {
            float a = bfc8[0];
            for (int j = 0; j < 32; j++) a += fh2[j] * Wfc8[j];
            out[512 + s] = sigmoidf_(a);
        }
        __syncthreads();
    }
}

// ---------------------------------------------------------------------------
extern "C" void kernel_launch(void* const* d_in, const int* in_sizes, int n_in,
                              void* d_out, int out_size, void* d_ws, size_t ws_size,
                              hipStream_t stream) {
    const float* frames  = (const float*)d_in[0];
    const float* W_spp   = (const float*)d_in[1];
    const float* b_spp   = (const float*)d_in[2];
    const float* W_fc7   = (const float*)d_in[3];
    const float* b_fc7   = (const float*)d_in[4];
    const float* W_fcast = (const float*)d_in[5];
    const float* b_fcast = (const float*)d_in[6];
    const float* Wih1    = (const float*)d_in[7];
    const float* Whh1    = (const float*)d_in[8];
    const float* bih1    = (const float*)d_in[9];
    const float* bhh1    = (const float*)d_in[10];
    const float* Wih2    = (const float*)d_in[11];
    const float* Whh2    = (const float*)d_in[12];
    const float* bih2    = (const float*)d_in[13];
    const float* bhh2    = (const float*)d_in[14];
    const float* Wfc8    = (const float*)d_in[15];
    const float* bfc8    = (const float*)d_in[16];
    float* out = (float*)d_out;

    // workspace layout (floats)
    float* ws     = (float*)d_ws;
    float* sppP   = ws;                   // 512*96    = 49152
    float* Wb1    = sppP + 49152;         // 96*4096   = 393216
    float* Wb7    = Wb1  + 393216;        // 4096*64   = 262144
    float* Wbc    = Wb7  + 262144;        // 4096*64   = 262144
    float* pooled = Wbc  + 262144;        // 512*4096  = 2097152
    float* emb    = pooled + 2097152;     // 512*64    = 32768
    float* fct    = emb  + 32768;         // 512*64    = 32768

    pad_transpose_wspp<<<(96 * 4096) / 256, 256, 0, stream>>>(W_spp, Wb1);
    transpose_w64<<<(4096 * 64) / 256, 256, 0, stream>>>(W_fc7, Wb7);
    transpose_w64<<<(4096 * 64) / 256, 256, 0, stream>>>(W_fcast, Wbc);

    spp_kernel<<<512 * 3, 160, 0, stream>>>(frames, sppP);

    // pooled = relu(sppP(512x96) @ Wb1(96x4096) + b_spp)
    gemm_relu_wmma<<<1024, 256, 0, stream>>>(sppP, Wb1, b_spp, pooled,
                                             /*tiles*/ 32 * 256, /*perRow*/ 256,
                                             /*K*/ 96, /*lda*/ 96, /*ldb*/ 4096, /*ldc*/ 4096);
    // embedded = relu(pooled @ Wb7 + b_fc7)
    gemm_relu_wmma<<<16, 256, 0, stream>>>(pooled, Wb7, b_fc7, emb,
                                           32 * 4, 4, 4096, 4096, 64, 64);
    // fcast = relu(pooled @ Wbc + b_fcast)
    gemm_relu_wmma<<<16, 256, 0, stream>>>(pooled, Wbc, b_fcast, fct,
                                           32 * 4, 4, 4096, 4096, 64, 64);

    lstm_kernel<<<1, 256, 0, stream>>>(emb, fct, Wih1, Whh1, bih1, bhh1,
                                       Wih2, Whh2, bih2, bhh2, Wfc8, bfc8, out);
}